// DINOMIMICClassification_50843822850059
// MI455X (gfx1250) — compile-verified
//
#include <hip/hip_runtime.h>
#include <hip/hip_bf16.h>

// Problem constants (match reference)
constexpr int E = 16;
constexpr int B = 512;
constexpr int D = 1536;
constexpr int H = 768;
constexpr int T = 2;

typedef __attribute__((ext_vector_type(2))) float v2f;
typedef __attribute__((ext_vector_type(8))) float v8f;

// ---------------------------------------------------------------------------
// Phase 0: stable bucketing of samples by expert (deterministic, no atomics).
// perm[off[e] .. off[e]+cnt[e]) = sample indices with head_idx == e, in order.
// ---------------------------------------------------------------------------
__global__ __launch_bounds__(32) void moe_bucket(const int* __restrict__ head,
                                                 int* __restrict__ cnt,
                                                 int* __restrict__ off,
                                                 int* __restrict__ perm) {
    __shared__ int scnt[E];
    __shared__ int soff[E];
    const int t = threadIdx.x;
    if (t < E) {
        int c = 0;
        for (int b = 0; b < B; ++b) c += (head[b] == t);
        scnt[t] = c;
    }
    __syncthreads();
    if (t == 0) {
        int a = 0;
        for (int e = 0; e < E; ++e) { soff[e] = a; a += scnt[e]; }
    }
    __syncthreads();
    if (t < E) {
        cnt[t] = scnt[t];
        off[t] = soff[t];
        int w = soff[t];
        for (int b = 0; b < B; ++b)
            if (head[b] == t) perm[w++] = b;
    }
}

// ---------------------------------------------------------------------------
// Layer 1: h1[p, :] = relu(x[perm[p], :] @ W1[e] + b1[e])   (K = D = 1536)
// One wave per 16x16 output tile; fp32 WMMA 16x16x4.
// A lane layout: lane -> row M = lane%16, K pair = 2*(lane/16)+{0,1}
// B lane layout: lane -> col N = lane%16, K rows  = 2*(lane/16)+{0,1}
// C/D layout:    vgpr r, lane -> (M = r + 8*(lane/16), N = lane%16)
// ---------------------------------------------------------------------------
__global__ __launch_bounds__(32) void moe_layer1(const float* __restrict__ x,
                                                 const float* __restrict__ W1,
                                                 const float* __restrict__ b1,
                                                 const int* __restrict__ cnt,
                                                 const int* __restrict__ off,
                                                 const int* __restrict__ perm,
                                                 float* __restrict__ h1) {
    const int e  = blockIdx.z;
    const int mt = blockIdx.y;
    const int n0 = blockIdx.x * 16;
    const int ne = cnt[e];
    if (mt * 16 >= ne) return;
    const int base = off[e];
    const int lane = threadIdx.x;
    const int half = lane >> 4;   // 0/1 -> K sub-pair / row sub-half
    const int l16  = lane & 15;

    // Gathered A row for this lane (clamp padding rows to a valid sample).
    int p  = mt * 16 + l16;
    int pc = p < ne ? p : ne - 1;
    const int src = perm[base + pc];
    const float* __restrict__ arow = x + (size_t)src * D + 2 * half;
    const float* __restrict__ bcol = W1 + (size_t)e * D * H + (size_t)(2 * half) * H + n0 + l16;

    v8f c = {};
#pragma unroll 4
    for (int k = 0; k < D; k += 4) {
        v2f a = *(const v2f*)(arow + k);           // K = k+2h, k+2h+1 (8B aligned)
        v2f b;
        b.x = bcol[(size_t)k * H];                 // W1[k+2h,     n]
        b.y = bcol[(size_t)k * H + H];             // W1[k+2h + 1, n]
        c = __builtin_amdgcn_wmma_f32_16x16x4_f32(false, a, false, b,
                                                  (short)0, c, false, false);
    }

    const float bias = b1[e * H + n0 + l16];
#pragma unroll
    for (int r = 0; r < 8; ++r) {
        const int M  = r + 8 * half;
        const int pr = mt * 16 + M;
        if (pr < ne) {
            float v = c[r] + bias;
            h1[(size_t)(base + pr) * H + n0 + l16] = v > 0.f ? v : 0.f;
        }
    }
}

// ---------------------------------------------------------------------------
// Layer 2: h2[p, :] = relu(h1[p, :] @ W2[e] + b2[e])   (K = H = 768)
// A rows are already gathered/contiguous (positions), no perm indirection.
// ---------------------------------------------------------------------------
__global__ __launch_bounds__(32) void moe_layer2(const float* __restrict__ h1,
                                                 const float* __restrict__ W2,
                                                 const float* __restrict__ b2,
                                                 const int* __restrict__ cnt,
                                                 const int* __restrict__ off,
                                                 float* __restrict__ h2) {
    const int e  = blockIdx.z;
    const int mt = blockIdx.y;
    const int n0 = blockIdx.x * 16;
    const int ne = cnt[e];
    if (mt * 16 >= ne) return;
    const int base = off[e];
    const int lane = threadIdx.x;
    const int half = lane >> 4;
    const int l16  = lane & 15;

    int p  = mt * 16 + l16;
    int pc = p < ne ? p : ne - 1;
    const float* __restrict__ arow = h1 + (size_t)(base + pc) * H + 2 * half;
    const float* __restrict__ bcol = W2 + (size_t)e * H * H + (size_t)(2 * half) * H + n0 + l16;

    v8f c = {};
#pragma unroll 4
    for (int k = 0; k < H; k += 4) {
        v2f a = *(const v2f*)(arow + k);
        v2f b;
        b.x = bcol[(size_t)k * H];
        b.y = bcol[(size_t)k * H + H];
        c = __builtin_amdgcn_wmma_f32_16x16x4_f32(false, a, false, b,
                                                  (short)0, c, false, false);
    }

    const float bias = b2[e * H + n0 + l16];
#pragma unroll
    for (int r = 0; r < 8; ++r) {
        const int M  = r + 8 * half;
        const int pr = mt * 16 + M;
        if (pr < ne) {
            float v = c[r] + bias;
            h2[(size_t)(base + pr) * H + n0 + l16] = v > 0.f ? v : 0.f;
        }
    }
}

// ---------------------------------------------------------------------------
// Layer 3: out[perm[p], t] = h2[p, :] . W3[e, :, t] + b3[e, t]   (T = 2)
// One 64-thread block per position; VALU dot + LDS tree reduce.
// ---------------------------------------------------------------------------
__global__ __launch_bounds__(64) void moe_layer3(const float* __restrict__ h2,
                                                 const float* __restrict__ W3,
                                                 const float* __restrict__ b3,
                                                 const int* __restrict__ head,
                                                 const int* __restrict__ perm,
                                                 float* __restrict__ out) {
    __shared__ float red0[64];
    __shared__ float red1[64];
    const int p = blockIdx.x;
    const int t = threadIdx.x;
    const int b = perm[p];
    const int e = head[b];
    const float* __restrict__ hrow = h2 + (size_t)p * H;
    const float* __restrict__ w    = W3 + (size_t)e * H * T;
    float s0 = 0.f, s1 = 0.f;
#pragma unroll 4
    for (int k = t; k < H; k += 64) {
        const float hv = hrow[k];
        s0 += hv * w[k * T + 0];
        s1 += hv * w[k * T + 1];
    }
    red0[t] = s0;
    red1[t] = s1;
    __syncthreads();
#pragma unroll
    for (int stride = 32; stride > 0; stride >>= 1) {
        if (t < stride) {
            red0[t] += red0[t + stride];
            red1[t] += red1[t + stride];
        }
        __syncthreads();
    }
    if (t == 0) {
        out[b * T + 0] = red0[0] + b3[e * T + 0];
        out[b * T + 1] = red1[0] + b3[e * T + 1];
    }
}

// ---------------------------------------------------------------------------
extern "C" void kernel_launch(void* const* d_in, const int* in_sizes, int n_in,
                              void* d_out, int out_size, void* d_ws, size_t ws_size,
                              hipStream_t stream) {
    (void)in_sizes; (void)n_in; (void)out_size; (void)ws_size;
    const float* x    = (const float*)d_in[0];
    const int*   head = (const int*)d_in[1];
    const float* W1   = (const float*)d_in[2];
    const float* b1   = (const float*)d_in[3];
    const float* W2   = (const float*)d_in[4];
    const float* b2   = (const float*)d_in[5];
    const float* W3   = (const float*)d_in[6];
    const float* b3   = (const float*)d_in[7];
    float* out = (float*)d_out;

    // Workspace layout (needs ~3.15 MB):
    //   [0)        int cnt[E]
    //   [64)       int off[E]
    //   [128)      int perm[B]
    //   [4096)     float h1[B*H]
    //   [+B*H*4)   float h2[B*H]
    char* ws   = (char*)d_ws;
    int*  cnt  = (int*)ws;
    int*  off  = cnt + E;
    int*  perm = off + E;
    float* h1  = (float*)(ws + 4096);
    float* h2  = h1 + (size_t)B * H;

    moe_bucket<<<1, 32, 0, stream>>>(head, cnt, off, perm);

    dim3 grid(H / 16, B / 16, E);   // 48 N-tiles, up to 32 M-tiles, 16 experts
    moe_layer1<<<grid, 32, 0, stream>>>(x, W1, b1, cnt, off, perm, h1);
    moe_layer2<<<grid, 32, 0, stream>>>(h1, W2, b2, cnt, off, h2);
    moe_layer3<<<B, 64, 0, stream>>>(h2, W3, b3, head, perm, out);
}